// SystemLayer_56873956933646
// MI455X (gfx1250) — compile-verified
//
#include <hip/hip_runtime.h>
#include <stdint.h>

// Problem constants (match reference)
#define B_ 16
#define N_ 8192
#define K_ 512
#define C_ 128
#define R_ (B_ * N_)                 // 131072 rows
// Output layout (flat floats, reference return order)
#define OUT_HA 0                     // hard_assign   [B*N]
#define OUT_PC (R_)                  // pred_classes  [B*N]
#define OUT_CB (2 * R_)              // comp_boxes    [B*K*4]
#define OUT_ID (2 * R_ + B_ * K_ * 4) // component_ids [B*K]

#define ROWS_PER_WAVE 8

// ---------------------------------------------------------------------------
// Init: comp_boxes accumulators (uint-ordered: +inf bits for min, 0 for max)
// and component_ids (= k, broadcast per batch). One thread per (b,k).
// ---------------------------------------------------------------------------
__global__ __launch_bounds__(256) void init_comp(float* __restrict__ out) {
  const int i = blockIdx.x * 256 + threadIdx.x;      // 0 .. B*K-1 (8192)
  unsigned* comp = (unsigned*)(out + OUT_CB);
  comp[4 * i + 0] = 0x7F800000u;  // +inf  (x1 min accumulator)
  comp[4 * i + 1] = 0x7F800000u;  // +inf  (y1 min accumulator)
  comp[4 * i + 2] = 0u;           // 0.0f  (x2 max accumulator)
  comp[4 * i + 3] = 0u;           // 0.0f  (y2 max accumulator)
  out[OUT_ID + i] = (float)(i & (K_ - 1));           // component id = k
}

// ---------------------------------------------------------------------------
// argmax over K=512 per row, streamed through LDS with async double-buffering.
// One wave (32 lanes) per block; each wave handles 8 rows of 2 KB.
// Fill: 4 x global_load_async_to_lds_b128 (32 lanes x 16B = 512B each),
// tagged th:TH_LOAD_NT: the 256 MB assign_probs tensor is single-use and
// larger than the 192 MB L2, so keep it evict-first and let L2 retain the
// 64 MB class_logits (re-hit on every graph replay) instead.
// While reducing buffer j from LDS, buffer j+1 is filling (ASYNCcnt pipelined).
// Lane 0 then scatters the row's box into the component accumulators with
// hardware u32 atomics (valid float ordering: boxes are non-negative).
// ---------------------------------------------------------------------------
__global__ __launch_bounds__(32) void argmax_k(const float* __restrict__ assign,
                                               const float* __restrict__ boxes,
                                               float* __restrict__ out) {
  __shared__ float lds[2 * K_];                      // 2 x 2KB row buffers
  const int lane = threadIdx.x;                      // 0..31 (wave32)
  const unsigned voff = (unsigned)(lane * 16);       // per-lane byte offset
  const unsigned lbase = (unsigned)(uintptr_t)&lds[0] + voff;
  unsigned* comp = (unsigned*)(out + OUT_CB);

  const int row0 = blockIdx.x * ROWS_PER_WAVE;

  // Prime buffer 0 with row0 (ASYNCcnt += 4)
  {
    const float* rp = assign + (size_t)row0 * K_;
    asm volatile(
        "global_load_async_to_lds_b128 %0, %1, %2 offset:0 th:TH_LOAD_NT\n\t"
        "global_load_async_to_lds_b128 %0, %1, %2 offset:512 th:TH_LOAD_NT\n\t"
        "global_load_async_to_lds_b128 %0, %1, %2 offset:1024 th:TH_LOAD_NT\n\t"
        "global_load_async_to_lds_b128 %0, %1, %2 offset:1536 th:TH_LOAD_NT"
        :: "v"(lbase), "v"(voff), "s"(rp)
        : "memory");
  }

  for (int j = 0; j < ROWS_PER_WAVE; ++j) {
    const int row = row0 + j;

    if (j + 1 < ROWS_PER_WAVE) {
      // Prefetch next row into the other buffer, then wait for current
      // buffer's 4 async loads (in-order completion => asynccnt <= 4).
      const float* rp = assign + (size_t)(row + 1) * K_;
      const unsigned lnext = lbase + (unsigned)(((j + 1) & 1) * 2048);
      asm volatile(
          "global_load_async_to_lds_b128 %0, %1, %2 offset:0 th:TH_LOAD_NT\n\t"
          "global_load_async_to_lds_b128 %0, %1, %2 offset:512 th:TH_LOAD_NT\n\t"
          "global_load_async_to_lds_b128 %0, %1, %2 offset:1024 th:TH_LOAD_NT\n\t"
          "global_load_async_to_lds_b128 %0, %1, %2 offset:1536 th:TH_LOAD_NT"
          :: "v"(lnext), "v"(voff), "s"(rp)
          : "memory");
      asm volatile("s_wait_asynccnt 0x4" ::: "memory");
    } else {
      asm volatile("s_wait_asynccnt 0x0" ::: "memory");
    }

    // Reduce current buffer: lane covers k = i*128 + lane*4 + c (increasing
    // within a lane => strict '>' keeps first occurrence, like jnp.argmax).
    const float4* lb = (const float4*)&lds[(j & 1) * K_];
    float bv = -__builtin_inff();
    int bi = 0;
#pragma unroll
    for (int i = 0; i < 4; ++i) {
      const float4 v = lb[lane + i * 32];            // ds_load_b128
      const int k0 = i * 128 + lane * 4;
      if (v.x > bv) { bv = v.x; bi = k0; }
      if (v.y > bv) { bv = v.y; bi = k0 + 1; }
      if (v.z > bv) { bv = v.z; bi = k0 + 2; }
      if (v.w > bv) { bv = v.w; bi = k0 + 3; }
    }
    // Wave32 tree reduction; on ties keep the smaller index (first occurrence)
#pragma unroll
    for (int off = 16; off >= 1; off >>= 1) {
      const float ov = __shfl_xor(bv, off, 32);
      const int   oi = __shfl_xor(bi, off, 32);
      if (ov > bv || (ov == bv && oi < bi)) { bv = ov; bi = oi; }
    }

    if (lane == 0) {
      out[OUT_HA + row] = (float)bi;
      const float4 bx = ((const float4*)boxes)[row];
      const int b = row >> 13;                       // row / N_
      unsigned* c = comp + 4 * ((size_t)b * K_ + bi);
      atomicMin(c + 0, __float_as_uint(bx.x));       // x1: segment min
      atomicMin(c + 1, __float_as_uint(bx.y));       // y1: segment min
      atomicMax(c + 2, __float_as_uint(bx.z));       // x2: segment max
      atomicMax(c + 3, __float_as_uint(bx.w));       // y2: segment max
    }
  }
}

// ---------------------------------------------------------------------------
// argmax over C=128 per row: one wave per row, one b128 load per lane.
// Default (RT) caching on purpose: 64 MB fits in the 192 MB L2 and is
// re-used across graph replays once the assign stream is non-temporal.
// ---------------------------------------------------------------------------
__global__ __launch_bounds__(256) void argmax_c(const float* __restrict__ logits,
                                                float* __restrict__ out) {
  const int t = blockIdx.x * 256 + threadIdx.x;
  const int row = t >> 5;
  const int lane = t & 31;
  const float4 v = ((const float4*)(logits + (size_t)row * C_))[lane];
  float bv = v.x;
  int bi = lane * 4;
  if (v.y > bv) { bv = v.y; bi = lane * 4 + 1; }
  if (v.z > bv) { bv = v.z; bi = lane * 4 + 2; }
  if (v.w > bv) { bv = v.w; bi = lane * 4 + 3; }
#pragma unroll
  for (int off = 16; off >= 1; off >>= 1) {
    const float ov = __shfl_xor(bv, off, 32);
    const int   oi = __shfl_xor(bi, off, 32);
    if (ov > bv || (ov == bv && oi < bi)) { bv = ov; bi = oi; }
  }
  if (lane == 0) out[OUT_PC + row] = (float)bi;
}

// ---------------------------------------------------------------------------
// Finalize: reinterpret accumulated uint bits as float, apply reference
// clamps (min with 1.0 covers empty segments that stayed +inf; max with 0.0
// matches the empty-segment default for x2/y2). One thread per scalar.
// ---------------------------------------------------------------------------
__global__ __launch_bounds__(256) void finalize_comp(float* __restrict__ out) {
  const int i = blockIdx.x * 256 + threadIdx.x;      // 0 .. B*K*4-1 (32768)
  const unsigned u = ((const unsigned*)(out + OUT_CB))[i];
  float v = __uint_as_float(u);
  v = ((i & 3) < 2) ? fminf(v, 1.0f) : fmaxf(v, 0.0f);
  out[OUT_CB + i] = v;
}

extern "C" void kernel_launch(void* const* d_in, const int* in_sizes, int n_in,
                              void* d_out, int out_size, void* d_ws, size_t ws_size,
                              hipStream_t stream) {
  const float* boxes  = (const float*)d_in[0];   // [B,N,4]
  const float* assign = (const float*)d_in[1];   // [B,N,K]
  const float* logits = (const float*)d_in[2];   // [B,N,C]
  float* out = (float*)d_out;

  init_comp<<<(B_ * K_) / 256, 256, 0, stream>>>(out);
  argmax_k<<<R_ / ROWS_PER_WAVE, 32, 0, stream>>>(assign, boxes, out);
  argmax_c<<<(R_ * 32) / 256, 256, 0, stream>>>(logits, out);
  finalize_comp<<<(B_ * K_ * 4) / 256, 256, 0, stream>>>(out);
}